// clip_nce_47158740910206
// MI455X (gfx1250) — compile-verified
//
#include <hip/hip_runtime.h>

// Problem sizes (fixed by the reference).
#define Q_N 16384
#define V_N 4096
#define TPB 256

// Fused-pass tiling: each block covers TR rows x TC cols, single HBM read.
#define TC 1024                      // columns per block (4 per thread)
#define NUM_CC (V_N / TC)            // 4 column chunks
#define TR 128                       // rows per block
#define NUM_RCK (Q_N / TR)           // 128 row chunks
#define RPC 4                        // rows per async pipeline chunk
#define NCHUNK (TR / RPC)            // 32 chunks

#ifndef __has_builtin
#define __has_builtin(x) 0
#endif

// Pointer types matching the ROCm builtin prototype:
//   __builtin_amdgcn_global_load_async_to_lds_b128(int4 AS1*, int4 AS3*, Ii, Ii)
typedef int v4i_t __attribute__((vector_size(4 * sizeof(int))));
typedef __attribute__((address_space(1))) v4i_t g_v4i;   // global
typedef __attribute__((address_space(3))) v4i_t l_v4i;   // LDS

#if defined(__HIP_DEVICE_COMPILE__)
  #if __has_builtin(__builtin_amdgcn_global_load_async_to_lds_b128) && \
      __has_builtin(__builtin_amdgcn_s_wait_asynccnt)
    #define HAVE_ASYNC_LDS 1
  #endif
#endif

#ifdef HAVE_ASYNC_LDS
  #define ASYNC_CP16(g, l) __builtin_amdgcn_global_load_async_to_lds_b128((g_v4i*)(g), (l_v4i*)(l), 0, 0)
  #define ASYNC_WAIT(n)    __builtin_amdgcn_s_wait_asynccnt(n)
#else  // host-pass parse / older toolchains: synchronous staging, waits are no-ops
  #define ASYNC_CP16(g, l) (*(float4*)(l) = *(const float4*)(g))
  #define ASYNC_WAIT(n)    ((void)0)
#endif

// ---------------- wave32 helpers ----------------
__device__ __forceinline__ float wave_sum(float v) {
  #pragma unroll
  for (int o = 16; o > 0; o >>= 1) v += __shfl_xor(v, o, 32);
  return v;
}

// Sequential (m,s) logsumexp-pair combine.
__device__ __forceinline__ void lse_combine(float& M, float& S, float pm, float ps) {
  if (pm > M) { S = S * __expf(M - pm) + ps; M = pm; }
  else        { S += ps * __expf(pm - M); }
}

// Monotone float<->uint encoding so integer atomicMax implements float max
// (order-independent => deterministic). enc(x) > 0 for all reals, so init 0 is
// an identity (every video has >= 1 query, so every slot gets updated).
__device__ __forceinline__ unsigned enc_f32(float f) {
  unsigned u = __float_as_uint(f);
  return (u & 0x80000000u) ? ~u : (u | 0x80000000u);
}
__device__ __forceinline__ float dec_f32(unsigned u) {
  return (u & 0x80000000u) ? __uint_as_float(u & 0x7FFFFFFFu) : __uint_as_float(~u);
}

// ---------------- fused single-read kernel ----------------
// Block (cc, rck): rows [rck*TR, +TR) x cols [cc*TC, +TC).
// Async double-buffered global->LDS staging; from each staged chunk we derive
//   * row partials: register 2-pass over 4 cols + wave32 shuffle LSE-combine,
//     per-(row,wave) results parked in LDS, merged once at the end;
//   * col partials: 4 per-thread online-LSE accumulators held in registers.
__global__ __launch_bounds__(TPB) void fused_lse_kernel(const float* __restrict__ scores,
                                                        float* __restrict__ rowPartM,
                                                        float* __restrict__ rowPartS,
                                                        float* __restrict__ colPartM,
                                                        float* __restrict__ colPartS) {
  __shared__ __align__(16) float buf[2][RPC * TC];   // 2 x 16KB
  __shared__ float rowWaveM[TR][TPB / 32];           // 4KB
  __shared__ float rowWaveS[TR][TPB / 32];           // 4KB

  const int t   = threadIdx.x;
  const int wv  = t >> 5;
  const int cc  = blockIdx.x;            // column chunk
  const int rck = blockIdx.y;            // row chunk
  const int c0  = cc * TC;
  const int r0  = rck * TR;
  const float* base = scores + (size_t)r0 * V_N + c0 + 4 * t;

  float cm[4], cs[4];
  #pragma unroll
  for (int k = 0; k < 4; ++k) { cm[k] = -INFINITY; cs[k] = 0.f; }

  auto issue = [&](int chunk, int b) {
    #pragma unroll
    for (int rr = 0; rr < RPC; ++rr) {
      const float* g = base + (size_t)(chunk * RPC + rr) * V_N;
      float* l = &buf[b][rr * TC + 4 * t];
      ASYNC_CP16(g, l);
    }
  };

  auto process = [&](int b, int chunk) {
    #pragma unroll
    for (int rr = 0; rr < RPC; ++rr) {
      const float4 xv = *(const float4*)&buf[b][rr * TC + 4 * t];
      float x[4] = {xv.x, xv.y, xv.z, xv.w};
      // row side: register 2-pass LSE over this thread's 4 columns
      float ml = fmaxf(fmaxf(x[0], x[1]), fmaxf(x[2], x[3]));
      float sl = __expf(x[0] - ml) + __expf(x[1] - ml) +
                 __expf(x[2] - ml) + __expf(x[3] - ml);
      // wave32 butterfly LSE-combine -> (m,s) over this wave's 128 columns
      #pragma unroll
      for (int o = 16; o > 0; o >>= 1) {
        const float mo = __shfl_xor(ml, o, 32);
        const float so = __shfl_xor(sl, o, 32);
        const float M  = fmaxf(ml, mo);
        sl = sl * __expf(ml - M) + so * __expf(mo - M);
        ml = M;
      }
      if ((t & 31) == 0) {
        rowWaveM[chunk * RPC + rr][wv] = ml;
        rowWaveS[chunk * RPC + rr][wv] = sl;
      }
      // column side: online LSE update of the 4 register accumulators
      #pragma unroll
      for (int k = 0; k < 4; ++k) {
        if (x[k] > cm[k]) { cs[k] = cs[k] * __expf(cm[k] - x[k]) + 1.f; cm[k] = x[k]; }
        else              { cs[k] += __expf(x[k] - cm[k]); }
      }
    }
  };

  // software pipeline: async engine fills buffer b^1 while we consume buffer b
  issue(0, 0);
  for (int chunk = 0; chunk < NCHUNK - 1; ++chunk) {
    const int cur = chunk & 1;
    issue(chunk + 1, cur ^ 1);
    ASYNC_WAIT(RPC);          // oldest RPC ops (buffer `cur`) complete, in-order
    __syncthreads();          // all waves' portions of `cur` visible
    process(cur, chunk);
    __syncthreads();          // everyone done reading `cur` before it is refilled
  }
  ASYNC_WAIT(0);
  __syncthreads();
  process((NCHUNK - 1) & 1, NCHUNK - 1);
  __syncthreads();            // rowWave* complete

  // merge per-wave row partials -> (m,s) per row for this column chunk
  if (t < TR) {
    float M = rowWaveM[t][0], S = rowWaveS[t][0];
    #pragma unroll
    for (int w = 1; w < TPB / 32; ++w) lse_combine(M, S, rowWaveM[t][w], rowWaveS[t][w]);
    rowPartM[(size_t)cc * Q_N + r0 + t] = M;   // [cc][row] -> coalesced
    rowPartS[(size_t)cc * Q_N + r0 + t] = S;
  }

  // column partials: [rck][col] -> coalesced float4 stores
  const float4 mv = make_float4(cm[0], cm[1], cm[2], cm[3]);
  const float4 sv = make_float4(cs[0], cs[1], cs[2], cs[3]);
  *(float4*)&colPartM[(size_t)rck * V_N + c0 + 4 * t] = mv;
  *(float4*)&colPartS[(size_t)rck * V_N + c0 + 4 * t] = sv;
}

// ---------------- row combine: 4 column-chunk partials -> t2v contribution ----------------
__global__ __launch_bounds__(TPB) void row_combine_kernel(const float* __restrict__ rowPartM,
                                                          const float* __restrict__ rowPartS,
                                                          const float* __restrict__ scores,
                                                          const int*   __restrict__ labels,
                                                          float* __restrict__ rowContrib) {
  const int q = blockIdx.x * TPB + threadIdx.x;
  float M = -INFINITY, S = 0.f;
  #pragma unroll
  for (int cc = 0; cc < NUM_CC; ++cc)
    lse_combine(M, S, rowPartM[(size_t)cc * Q_N + q], rowPartS[(size_t)cc * Q_N + q]);
  const float diag = scores[(size_t)q * V_N + labels[q]];
  rowContrib[q] = (M + __logf(S)) - diag;
}

// ---------------- segment-LSE of diagonal scores over labels ----------------
__global__ void seg_init_kernel(unsigned* __restrict__ segMu, float* __restrict__ segS) {
  const int v = blockIdx.x * blockDim.x + threadIdx.x;
  if (v < V_N) { segMu[v] = 0u; segS[v] = 0.f; }
}

__global__ void seg_max_kernel(const float* __restrict__ scores, const int* __restrict__ labels,
                               unsigned* __restrict__ segMu) {
  const int q = blockIdx.x * blockDim.x + threadIdx.x;
  if (q < Q_N) {
    const int v = labels[q];
    atomicMax(&segMu[v], enc_f32(scores[(size_t)q * V_N + v]));
  }
}

__global__ void seg_sum_kernel(const float* __restrict__ scores, const int* __restrict__ labels,
                               const unsigned* __restrict__ segMu, float* __restrict__ segS) {
  const int q = blockIdx.x * blockDim.x + threadIdx.x;
  if (q < Q_N) {
    const int v = labels[q];
    const float d = scores[(size_t)q * V_N + v];
    atomicAdd(&segS[v], __expf(d - dec_f32(segMu[v])));
  }
}

// ---------------- column combine: 128 row-chunk partials + nominator ----------------
__global__ __launch_bounds__(TPB) void col_finalize_kernel(const float* __restrict__ colPartM,
                                                           const float* __restrict__ colPartS,
                                                           const unsigned* __restrict__ segMu,
                                                           const float* __restrict__ segS,
                                                           float* __restrict__ colContrib) {
  const int col = blockIdx.x * TPB + threadIdx.x;
  float M = -INFINITY, S = 0.f;
  #pragma unroll 4
  for (int k = 0; k < NUM_RCK; ++k)
    lse_combine(M, S, colPartM[(size_t)k * V_N + col], colPartS[(size_t)k * V_N + col]);
  const float colLSE = M + __logf(S);
  const float nom    = dec_f32(segMu[col]) + __logf(segS[col]);
  colContrib[col] = colLSE - nom;
}

// ---------------- deterministic final scalar reduce ----------------
__global__ __launch_bounds__(TPB) void final_reduce_kernel(const float* __restrict__ rowContrib,
                                                           const float* __restrict__ colContrib,
                                                           float* __restrict__ out) {
  __shared__ float red[TPB / 32];
  float a = 0.f, b = 0.f;
  for (int i = threadIdx.x; i < Q_N; i += TPB) a += rowContrib[i];
  for (int i = threadIdx.x; i < V_N; i += TPB) b += colContrib[i];
  float x = a * (1.0f / Q_N) + b * (1.0f / V_N);
  x = wave_sum(x);
  if ((threadIdx.x & 31) == 0) red[threadIdx.x >> 5] = x;
  __syncthreads();
  if (threadIdx.x == 0) {
    float tot = red[0];
    #pragma unroll
    for (int w = 1; w < TPB / 32; ++w) tot += red[w];
    out[0] = tot;
  }
}

extern "C" void kernel_launch(void* const* d_in, const int* in_sizes, int n_in,
                              void* d_out, int out_size, void* d_ws, size_t ws_size,
                              hipStream_t stream) {
  const float* scores = (const float*)d_in[0];   // [Q_N, V_N] f32
  const int*   labels = (const int*)d_in[1];     // [Q_N] i32
  float* out = (float*)d_out;                    // scalar f32

  // workspace layout (floats), ~4.8 MB total:
  float*    ws         = (float*)d_ws;
  float*    rowContrib = ws;                               // Q_N
  float*    colContrib = rowContrib + Q_N;                 // V_N
  unsigned* segMu      = (unsigned*)(colContrib + V_N);    // V_N
  float*    segS       = (float*)(segMu + V_N);            // V_N
  float*    rowPartM   = segS + V_N;                       // NUM_CC * Q_N
  float*    rowPartS   = rowPartM + (size_t)NUM_CC * Q_N;  // NUM_CC * Q_N
  float*    colPartM   = rowPartS + (size_t)NUM_CC * Q_N;  // NUM_RCK * V_N
  float*    colPartS   = colPartM + (size_t)NUM_RCK * V_N; // NUM_RCK * V_N

  seg_init_kernel<<<V_N / TPB, TPB, 0, stream>>>(segMu, segS);
  seg_max_kernel<<<Q_N / TPB, TPB, 0, stream>>>(scores, labels, segMu);
  seg_sum_kernel<<<Q_N / TPB, TPB, 0, stream>>>(scores, labels, segMu, segS);

  fused_lse_kernel<<<dim3(NUM_CC, NUM_RCK), TPB, 0, stream>>>(scores, rowPartM, rowPartS,
                                                              colPartM, colPartS);
  row_combine_kernel<<<Q_N / TPB, TPB, 0, stream>>>(rowPartM, rowPartS, scores, labels, rowContrib);
  col_finalize_kernel<<<V_N / TPB, TPB, 0, stream>>>(colPartM, colPartS, segMu, segS, colContrib);
  final_reduce_kernel<<<1, TPB, 0, stream>>>(rowContrib, colContrib, out);
}